// Model_DNN_2783138807866
// MI455X (gfx1250) — compile-verified
//
#include <hip/hip_runtime.h>
#include <hip/hip_bf16.h>

// CDNA5 (gfx1250) wave32 implementation of:
//   out[b,:] = sum_{o=1..5} tanh( (h[b,:]^o) @ W[b,:,:] )
// Memory-bound (256 MB of f32 W streamed once, ~2.5 FLOP/byte), so compute in
// exact f32 via V_WMMA_F32_16X16X4_F32; all W traffic is non-temporal.

typedef __attribute__((ext_vector_type(2))) float v2f;
typedef __attribute__((ext_vector_type(8))) float v8f;

#define EMB 64        // D
#define ORDERS 5
#define WAVES_PER_BLOCK 8
#define THREADS_PER_BLOCK (WAVES_PER_BLOCK * 32)

// x^(m+1) for m in [0,4], 0 otherwise (rows 5..15 of the A tile are zero).
__device__ __forceinline__ float pow_sel(float x, int m) {
    float p1 = x;
    float p2 = p1 * x;
    float p3 = p2 * x;
    float p4 = p3 * x;
    float p5 = p4 * x;
    float r = 0.0f;
    r = (m == 0) ? p1 : r;
    r = (m == 1) ? p2 : r;
    r = (m == 2) ? p3 : r;
    r = (m == 3) ? p4 : r;
    r = (m == 4) ? p5 : r;
    return r;
}

__global__ __launch_bounds__(THREADS_PER_BLOCK)
void poly_tanh_wmma_kernel(const float* __restrict__ his,   // [B,64]
                           const float* __restrict__ ad,    // [B,64*64] row-major W[d][e]
                           float* __restrict__ out,         // [B,64]
                           int nbatch) {
    const int lane = threadIdx.x & 31;
    const int wave = threadIdx.x >> 5;
    const int e = blockIdx.x * WAVES_PER_BLOCK + wave;   // example index (wave-uniform)
    if (e >= nbatch) return;                              // uniform per wave: EXEC stays all-1s

    const float* __restrict__ h = his + (size_t)e * EMB;
    const float* __restrict__ W = ad  + (size_t)e * (EMB * EMB);

    const int m     = lane & 15;              // A-tile row (order index)
    const int n     = lane & 15;              // B/C/D column within 16-wide N tile
    const int khalf = (lane < 16) ? 0 : 2;    // K sub-offset per ISA A/B VGPR layout

    v8f acc[4];
    #pragma unroll
    for (int nt = 0; nt < 4; ++nt)
        acc[nt] = (v8f){0.f, 0.f, 0.f, 0.f, 0.f, 0.f, 0.f, 0.f};

    // K = 64 split into 16 steps of 4; N = 64 split into 4 tiles of 16.
    #pragma unroll 8
    for (int kt = 0; kt < 16; ++kt) {
        const int kb = kt * 4 + khalf;

        // A tile (16x4 f32): lane holds A[m][kb] and A[m][kb+1].
        // kb is even -> 8-byte aligned -> single global_load_b64.
        const v2f hx = *(const v2f*)(h + kb);  // broadcast within half-wave
        v2f a;
        a.x = pow_sel(hx.x, m);
        a.y = pow_sel(hx.y, m);

        // B tiles (4x16 f32 each): lane holds W[kb][col] and W[kb+1][col].
        const float* __restrict__ wrow = W + (size_t)kb * EMB + n;
        #pragma unroll
        for (int nt = 0; nt < 4; ++nt) {
            v2f b;
            b.x = __builtin_nontemporal_load(wrow + nt * 16);        // row kb
            b.y = __builtin_nontemporal_load(wrow + nt * 16 + EMB);  // row kb+1
            // D = A*B + C, exact f32:  v_wmma_f32_16x16x4_f32
            acc[nt] = __builtin_amdgcn_wmma_f32_16x16x4_f32(
                /*neg_a=*/false, a, /*neg_b=*/false, b,
                /*c_mod=*/(short)0, acc[nt],
                /*reuse_a=*/false, /*reuse_b=*/false);
        }
    }

    // Epilogue: rows M=0..4 (the 5 orders) live in C VGPRs 0..4 of lanes 0-15.
    // Rows 5..15 are exactly zero (A rows were zero), so they never contribute.
    if (lane < 16) {
        float* __restrict__ o = out + (size_t)e * EMB + lane;
        #pragma unroll
        for (int nt = 0; nt < 4; ++nt) {
            float s = tanhf(acc[nt][0]) + tanhf(acc[nt][1]) + tanhf(acc[nt][2])
                    + tanhf(acc[nt][3]) + tanhf(acc[nt][4]);
            __builtin_nontemporal_store(s, o + nt * 16);
        }
    }
}

extern "C" void kernel_launch(void* const* d_in, const int* in_sizes, int n_in,
                              void* d_out, int out_size, void* d_ws, size_t ws_size,
                              hipStream_t stream) {
    const float* his = (const float*)d_in[0];   // [B,64] f32
    const float* ad  = (const float*)d_in[1];   // [B,4096] f32
    float* out = (float*)d_out;                 // [B,64] f32

    const int nbatch = in_sizes[0] / EMB;       // B = 16384
    const int blocks = (nbatch + WAVES_PER_BLOCK - 1) / WAVES_PER_BLOCK;

    poly_tanh_wmma_kernel<<<blocks, THREADS_PER_BLOCK, 0, stream>>>(his, ad, out, nbatch);
}